// GCN_77369540870414
// MI455X (gfx1250) — compile-verified
//
#include <hip/hip_runtime.h>

typedef __attribute__((ext_vector_type(2))) float v2f;
typedef __attribute__((ext_vector_type(8))) float v8f;

#define GCN_N_NODES 100000
#define GCN_N_EDGES 1600000
#define GCN_FEAT 128

// ---------------------------------------------------------------------------
// Zero-fill (float4 granularity).
// ---------------------------------------------------------------------------
__global__ __launch_bounds__(256) void gcn_zero_kernel(float4* __restrict__ p, int n4) {
  int i = blockIdx.x * 256 + threadIdx.x;
  if (i < n4) p[i] = make_float4(0.f, 0.f, 0.f, 0.f);
}

// ---------------------------------------------------------------------------
// Edge scatter-add: accum[dst[e], :] += h[src[e], :].
// One wave32 per edge; each lane moves one float4 (32 x 4 = 128 feats).
// Atomic f32 adds at AGENT scope -> hardware global_atomic_add_f32; the 51 MB
// accumulator is L2-resident on MI455X (192 MB L2), so atomics stay on-chip.
// ---------------------------------------------------------------------------
__global__ __launch_bounds__(256) void gcn_scatter_add_kernel(
    const float* __restrict__ h, const int* __restrict__ src,
    const int* __restrict__ dst, float* __restrict__ accum) {
  long long gid = (long long)blockIdx.x * 256 + threadIdx.x;
  int e    = (int)(gid >> 5);
  int lane = (int)(gid & 31);
  if (e >= GCN_N_EDGES) return;
  int s = src[e];
  int d = dst[e];
  const float4 v = reinterpret_cast<const float4*>(h + (size_t)s * GCN_FEAT)[lane];
  float* p = accum + (size_t)d * GCN_FEAT + (size_t)lane * 4;
  __hip_atomic_fetch_add(p + 0, v.x, __ATOMIC_RELAXED, __HIP_MEMORY_SCOPE_AGENT);
  __hip_atomic_fetch_add(p + 1, v.y, __ATOMIC_RELAXED, __HIP_MEMORY_SCOPE_AGENT);
  __hip_atomic_fetch_add(p + 2, v.z, __ATOMIC_RELAXED, __HIP_MEMORY_SCOPE_AGENT);
  __hip_atomic_fetch_add(p + 3, v.w, __ATOMIC_RELAXED, __HIP_MEMORY_SCOPE_AGENT);
}

// ---------------------------------------------------------------------------
// out[M,N] = act(A[M,128] @ W[128,N] + bias), fp32 WMMA 16x16x4.
// N is a compile-time constant: B loads become global_load_b32 with immediate
// offsets (k*N*4 <= 63.5 KB fits the signed 24-bit IOFFSET), and for N % 16
// == 0 all column guards fold away. For N=40 the guard is a branchless
// v_cndmask select on a clamped in-bounds load -- no EXEC divergence inside
// the WMMA loop (EXEC must be all-1s around WMMA anyway).
//
// Block: 256 threads = 8 waves; one 16-row M tile per block, one 16-col N tile
// per wave. A tile staged in LDS with stride 132 (132 % 64 = 4 -> the 16
// same-column fragment reads hit 16 distinct banks; pairs read as ds_load_b64).
//
// V_WMMA_F32_16X16X4_F32 operand layout (ISA 7.12.2):
//   A (16x4): lanes 0-15 row M=lane, v[0]=K0,v[1]=K1; lanes 16-31 K2,K3.
//   B (4x16): lanes 0-15 col N=lane, v[0]=K0,v[1]=K1; lanes 16-31 K2,K3.
//   C/D (16x16): VGPR r, lanes 0-15 -> (M=r, N=lane); lanes 16-31 -> (M=r+8).
// ---------------------------------------------------------------------------
template <int N, bool RELU>
__global__ __launch_bounds__(256) void gcn_gemm_bias_relu_kernel(
    const float* __restrict__ A,     // [M x 128]
    const float* __restrict__ W,     // [128 x N]
    const float* __restrict__ bias,  // [N]
    float* __restrict__ out) {       // [M x N]
  constexpr int K = GCN_FEAT;
  constexpr int LDA = 132;
  constexpr int NTILES = (N + 15) >> 4;
  constexpr bool FULL = (N % 16) == 0;
  __shared__ float sA[16 * LDA];

  const int m0  = blockIdx.x * 16;
  const int tid = threadIdx.x;

  // Cooperative A-tile load: 16x128 floats = 512 float4, 2 per thread.
  const float4* Ag = reinterpret_cast<const float4*>(A + (size_t)m0 * K);
#pragma unroll
  for (int i = 0; i < 2; ++i) {
    int idx4 = tid + i * 256;   // 0..511
    int row  = idx4 >> 5;       // 32 float4 per row
    int c4   = idx4 & 31;
    float4 v = Ag[idx4];
    float* sp = &sA[row * LDA + c4 * 4];
    sp[0] = v.x; sp[1] = v.y; sp[2] = v.z; sp[3] = v.w;
  }
  __syncthreads();

  const int wave = tid >> 5;
  const int lane = tid & 31;
  if (wave >= NTILES) return;  // wave-uniform; EXEC stays all-1s in live waves

  const int n0    = wave * 16;
  const int col   = n0 + (lane & 15);
  const int khalf = (lane >> 4) << 1;   // 0 for lanes 0-15, 2 for lanes 16-31
  const int arow  = lane & 15;
  const bool colok = FULL ? true : (col < N);
  const int colc  = FULL ? col : (colok ? col : 0);  // clamped, always in range

  // Per-lane base pointers; all loop offsets become immediates after unroll.
  const float* ap = &sA[arow * LDA + khalf];
  const float* bp = W + (size_t)khalf * N + colc;

  v8f acc = {};
#pragma unroll
  for (int k = 0; k < K; k += 4) {
    v2f a, b;
    a.x = ap[k + 0];          // ds_load_b64 (contiguous pair)
    a.y = ap[k + 1];
    float bx = bp[(size_t)k * N];        // global_load_b32, imm offset
    float by = bp[(size_t)k * N + N];    // global_load_b32, imm offset
    b.x = colok ? bx : 0.f;   // v_cndmask (folds away when FULL)
    b.y = colok ? by : 0.f;
    acc = __builtin_amdgcn_wmma_f32_16x16x4_f32(
        /*neg_a=*/false, a, /*neg_b=*/false, b,
        /*c_mod=*/(short)0, acc, /*reuse_a=*/false, /*reuse_b=*/false);
  }

  const int rowbase = m0 + ((lane >> 4) << 3);
  const float bv = colok ? bias[colc] : 0.f;
#pragma unroll
  for (int r = 0; r < 8; ++r) {
    float v = acc[r] + bv;
    if (RELU) v = fmaxf(v, 0.f);
    if (colok) out[(size_t)(rowbase + r) * N + col] = v;
  }
}

// ---------------------------------------------------------------------------
// Host-side launch: 3 GCN layers, ping-pong buffers in workspace.
// ---------------------------------------------------------------------------
extern "C" void kernel_launch(void* const* d_in, const int* in_sizes, int n_in,
                              void* d_out, int out_size, void* d_ws, size_t ws_size,
                              hipStream_t stream) {
  (void)in_sizes; (void)n_in; (void)out_size; (void)ws_size;
  const float* features = (const float*)d_in[0];
  const int*   src      = (const int*)d_in[1];
  const int*   dst      = (const int*)d_in[2];
  const float* W1       = (const float*)d_in[3];
  const float* b1       = (const float*)d_in[4];
  const float* W2       = (const float*)d_in[5];
  const float* b2       = (const float*)d_in[6];
  const float* W3       = (const float*)d_in[7];
  const float* b3       = (const float*)d_in[8];
  float* outp = (float*)d_out;

  float* bufA = (float*)d_ws;                          // aggregation accumulator
  float* bufB = bufA + (size_t)GCN_N_NODES * GCN_FEAT; // hidden activations

  const int n4 = GCN_N_NODES * GCN_FEAT / 4;           // 3.2M float4
  dim3 blk(256);
  dim3 zgrid((n4 + 255) / 256);
  dim3 sgrid(((long long)GCN_N_EDGES * 32) / 256);     // 200000 blocks
  dim3 ggrid(GCN_N_NODES / 16);                        // 6250 blocks (exact)

  // Layer 1: aggregate features -> GEMM W1 + b1 + ReLU
  gcn_zero_kernel<<<zgrid, blk, 0, stream>>>((float4*)bufA, n4);
  gcn_scatter_add_kernel<<<sgrid, blk, 0, stream>>>(features, src, dst, bufA);
  gcn_gemm_bias_relu_kernel<128, true><<<ggrid, blk, 0, stream>>>(bufA, W1, b1, bufB);

  // Layer 2
  gcn_zero_kernel<<<zgrid, blk, 0, stream>>>((float4*)bufA, n4);
  gcn_scatter_add_kernel<<<sgrid, blk, 0, stream>>>(bufB, src, dst, bufA);
  gcn_gemm_bias_relu_kernel<128, true><<<ggrid, blk, 0, stream>>>(bufA, W2, b2, bufB);

  // Layer 3 (no ReLU, N = 40 -> 3 N-tiles, branchless column masking)
  gcn_zero_kernel<<<zgrid, blk, 0, stream>>>((float4*)bufA, n4);
  gcn_scatter_add_kernel<<<sgrid, blk, 0, stream>>>(bufB, src, dst, bufA);
  gcn_gemm_bias_relu_kernel<40, false><<<ggrid, blk, 0, stream>>>(bufA, W3, b3, outp);
}